// GraphSAGENet_2310692405679
// MI455X (gfx1250) — compile-verified
//
#include <hip/hip_runtime.h>

#define N_NODES 100000
#define CH      128
#define N_EDGES 1600000

typedef float v2f __attribute__((ext_vector_type(2)));
typedef float v8f __attribute__((ext_vector_type(8)));

// ---------------------------------------------------------------- utilities
__global__ void sage_zero_kernel(float* __restrict__ p, int n) {
    int i = blockIdx.x * 256 + threadIdx.x;
    if (i < n) p[i] = 0.0f;
}

// degree: deg[dst] += 1
__global__ void sage_deg_kernel(const long long* __restrict__ dst,
                                float* __restrict__ deg) {
    int e = blockIdx.x * 256 + threadIdx.x;
    if (e < N_EDGES) atomicAdd(&deg[(int)dst[e]], 1.0f);
}

// in-place: deg -> 1/max(deg,1)
__global__ void sage_invdeg_kernel(float* __restrict__ deg) {
    int i = blockIdx.x * 256 + threadIdx.x;
    if (i < N_NODES) deg[i] = 1.0f / fmaxf(deg[i], 1.0f);
}

// scatter-add: agg[dst] += feat[src]; 4 threads per edge, 32 channels each.
// Gather working set (51.2 MB) and target (51.2 MB) both fit in 192 MB L2.
__global__ void sage_scatter_kernel(const long long* __restrict__ src,
                                    const long long* __restrict__ dst,
                                    const float* __restrict__ feat,
                                    float* __restrict__ agg) {
    int t = blockIdx.x * 256 + threadIdx.x;
    int e = t >> 2;
    if (e >= N_EDGES) return;
    int part = (t & 3) * 32;
    int s = (int)src[e];
    int d = (int)dst[e];
    const float4* xs = (const float4*)(feat + (size_t)s * CH + part);
    float* ag = agg + (size_t)d * CH + part;
#pragma unroll
    for (int i = 0; i < 8; ++i) {
        float4 v = xs[i];
        atomicAdd(ag + i * 4 + 0, v.x);
        atomicAdd(ag + i * 4 + 1, v.y);
        atomicAdd(ag + i * 4 + 2, v.z);
        atomicAdd(ag + i * 4 + 3, v.w);
    }
}

// Fused GEMM: out = relu( (A1 * invdeg[row]) @ W1  +  A2 @ W2  + bias )
// A1,A2: [N_NODES,128] row-major; W1,W2: [128,128] row-major (k,n); out: [N_NODES,128].
// One wave computes a 16-row x 128-col slab using V_WMMA_F32_16X16X4_F32.
// Block = 256 threads = 8 waves = 128 rows per block.
//
// VGPR layouts (ISA 7.12.2, wave32):
//  A 16x4 f32: lane l<16 holds A[l][k0+0],A[l][k0+1]; lane l>=16 holds A[l-16][k0+2],A[l-16][k0+3]
//  B 4x16 f32: lane l<16 holds B[k0+0][l],B[k0+1][l]; lane l>=16 holds B[k0+2][l-16],B[k0+3][l-16]
//  C/D 16x16:  lane l<16: vgpr j = D[j][l];  lane l>=16: vgpr j = D[8+j][l-16]
__global__ void __launch_bounds__(256)
sage_gemm_kernel(const float* __restrict__ A1,      // aggregated neighbor sums
                 const float* __restrict__ invdeg,  // per-row scale for A1
                 const float* __restrict__ A2,      // root features
                 const float* __restrict__ W1,      // lin_l weight [k][n]
                 const float* __restrict__ W2,      // lin_r weight [k][n]
                 const float* __restrict__ bias,    // [128]
                 float* __restrict__ out) {
    const int wave = threadIdx.x >> 5;
    const int lane = threadIdx.x & 31;
    const int row0 = blockIdx.x * 128 + wave * 16;
    if (row0 >= N_NODES) return;   // wave-uniform: EXEC stays all-1s for WMMA

    const int r     = lane & 15;        // A row within tile == B/D column within tile
    const int khalf = (lane >> 4) * 2;  // 0 for lanes 0-15, 2 for lanes 16-31
    const int row   = row0 + r;

    const float scale = invdeg[row];
    const float* a1p = A1 + (size_t)row * CH + khalf;
    const float* a2p = A2 + (size_t)row * CH + khalf;
    const float* w1p = W1 + (size_t)khalf * CH + r;   // [k0*CH + n*16] indexes B elems
    const float* w2p = W2 + (size_t)khalf * CH + r;

    v8f acc[8];
#pragma unroll
    for (int n = 0; n < 8; ++n) acc[n] = (v8f)(0.0f);

    // ---- mean-neighbor term: (A1 * scale) @ W1 ----
    for (int k0 = 0; k0 < CH; k0 += 4) {
        v2f a;
        a.x = a1p[k0] * scale;
        a.y = a1p[k0 + 1] * scale;
        const float* wb = w1p + (size_t)k0 * CH;
#pragma unroll
        for (int n = 0; n < 8; ++n) {
            v2f b;
            b.x = wb[n * 16];
            b.y = wb[CH + n * 16];
            acc[n] = __builtin_amdgcn_wmma_f32_16x16x4_f32(
                false, a, false, b, (short)0, acc[n], false, false);
        }
    }

    // ---- root term: A2 @ W2 (same accumulators) ----
    for (int k0 = 0; k0 < CH; k0 += 4) {
        v2f a;
        a.x = a2p[k0];
        a.y = a2p[k0 + 1];
        const float* wb = w2p + (size_t)k0 * CH;
#pragma unroll
        for (int n = 0; n < 8; ++n) {
            v2f b;
            b.x = wb[n * 16];
            b.y = wb[CH + n * 16];
            acc[n] = __builtin_amdgcn_wmma_f32_16x16x4_f32(
                false, a, false, b, (short)0, acc[n], false, false);
        }
    }

    // ---- epilogue: bias + ReLU, scatter D tile to memory ----
    const int mBase = row0 + (lane >> 4) * 8;   // lanes 0-15 -> rows 0..7, 16-31 -> rows 8..15
#pragma unroll
    for (int n = 0; n < 8; ++n) {
        const int c = n * 16 + r;
        const float bv = bias[c];
#pragma unroll
        for (int j = 0; j < 8; ++j) {
            float v = acc[n][j] + bv;
            out[(size_t)(mBase + j) * CH + c] = fmaxf(v, 0.0f);
        }
    }
}

// ---------------------------------------------------------------- launcher
extern "C" void kernel_launch(void* const* d_in, const int* in_sizes, int n_in,
                              void* d_out, int out_size, void* d_ws, size_t ws_size,
                              hipStream_t stream) {
    (void)in_sizes; (void)n_in; (void)out_size; (void)ws_size;

    const float*     x    = (const float*)d_in[0];
    const long long* edge = (const long long*)d_in[1];   // [2, E] int64
    const float*     Wl1  = (const float*)d_in[2];
    const float*     Wr1  = (const float*)d_in[3];
    const float*     b1   = (const float*)d_in[4];
    const float*     Wl2  = (const float*)d_in[5];
    const float*     Wr2  = (const float*)d_in[6];
    const float*     b2   = (const float*)d_in[7];
    float*           out  = (float*)d_out;

    const long long* src = edge;
    const long long* dst = edge + N_EDGES;

    // workspace layout: agg [N*CH] | h [N*CH] | invdeg [N]
    float* agg  = (float*)d_ws;
    float* h    = agg + (size_t)N_NODES * CH;
    float* ideg = h + (size_t)N_NODES * CH;

    const int featN = N_NODES * CH;                      // 12.8M
    const int zeroGridF = (featN + 255) / 256;
    const int zeroGridN = (N_NODES + 255) / 256;
    const int edgeGrid  = (N_EDGES + 255) / 256;
    const int scatGrid  = (N_EDGES * 4 + 255) / 256;
    const int gemmGrid  = (N_NODES + 127) / 128;

    // degree -> inv_deg
    sage_zero_kernel<<<zeroGridN, 256, 0, stream>>>(ideg, N_NODES);
    sage_deg_kernel<<<edgeGrid, 256, 0, stream>>>(dst, ideg);
    sage_invdeg_kernel<<<zeroGridN, 256, 0, stream>>>(ideg);

    // layer 1
    sage_zero_kernel<<<zeroGridF, 256, 0, stream>>>(agg, featN);
    sage_scatter_kernel<<<scatGrid, 256, 0, stream>>>(src, dst, x, agg);
    sage_gemm_kernel<<<gemmGrid, 256, 0, stream>>>(agg, ideg, x, Wl1, Wr1, b1, h);

    // layer 2
    sage_zero_kernel<<<zeroGridF, 256, 0, stream>>>(agg, featN);
    sage_scatter_kernel<<<scatGrid, 256, 0, stream>>>(src, dst, h, agg);
    sage_gemm_kernel<<<gemmGrid, 256, 0, stream>>>(agg, ideg, h, Wl2, Wr2, b2, out);
}